// RankingLoss_57904749084751
// MI455X (gfx1250) — compile-verified
//
#include <hip/hip_runtime.h>

#define MARGIN_F   0.1f
#define N          2048
#define JSPLIT     8
#define JCHUNK     (N / JSPLIT)   // 256
#define BLK        256
#define NWAVES     (BLK / 32)     // 8 wave32s
#define ITILES     (N / 16)       // 128
#define NEG_SENT  (-1.0e30f)

typedef __attribute__((ext_vector_type(2))) float v2f;
typedef __attribute__((ext_vector_type(8))) float v8f;

// One block = (row b, j-chunk js). Computes the partial masked hinge sum for
// that row over its j range using V_WMMA_F32_16X16X4_F32 as an exact f32
// tile-row-sum engine (B = ones).
__global__ void __launch_bounds__(BLK)
rank_rows_kernel(const float* __restrict__ scores, const int* __restrict__ labels,
                 float* __restrict__ partial, float* __restrict__ npairs)
{
    __shared__ float tpos[N];     // pos_i ? margin - s_i : -1e30
    __shared__ float sneg[N];     // neg_j ? s_j        : -1e30
    __shared__ float wsum[NWAVES];
    __shared__ int   cpos_sh;

    const int b   = blockIdx.x;
    const int js  = blockIdx.y;
    const int tid = threadIdx.x;

    if (tid == 0) cpos_sh = 0;
    __syncthreads();

    int mypos = 0;
    for (int k = tid; k < N; k += BLK) {
        float s = scores[b * N + k];
        int   p = labels[b * N + k];           // labels in {0,1}; neg == !pos
        tpos[k] = p ? (MARGIN_F - s) : NEG_SENT;
        sneg[k] = p ? NEG_SENT       : s;
        mypos  += p;
    }
    atomicAdd(&cpos_sh, mypos);
    __syncthreads();

    const int lane = tid & 31;
    const int wave = tid >> 5;
    const int m    = lane & 15;           // A-matrix row (M) this lane feeds
    const int k0   = (lane >> 4) * 2;     // K base: lanes 0-15 -> K0/K1, 16-31 -> K2/K3

    v2f ones; ones[0] = 1.0f; ones[1] = 1.0f;
    v8f acc[4];
    #pragma unroll
    for (int u = 0; u < 4; ++u) acc[u] = (v8f){};

    const int jbase = js * JCHUNK;
    for (int it = wave; it < ITILES; it += NWAVES) {
        const float t0 = tpos[it * 16 + m];
        for (int j = jbase; j < jbase + JCHUNK; j += 16) {
            #pragma unroll
            for (int u = 0; u < 4; ++u) {   // 4 independent WMMA chains
                float2 sj = *reinterpret_cast<const float2*>(&sneg[j + 4 * u + k0]);
                v2f a;
                a[0] = fmaxf(t0 + sj.x, 0.0f);
                a[1] = fmaxf(t0 + sj.y, 0.0f);
                acc[u] = __builtin_amdgcn_wmma_f32_16x16x4_f32(
                    false, a, false, ones, (short)0, acc[u], false, false);
            }
        }
    }

    // All 16 columns of D are identical row-sums. VGPR r holds M=r (lanes 0-15)
    // and M=8+r (lanes 16-31): fold the 8 VGPRs, then fold the two halves.
    float s = 0.0f;
    #pragma unroll
    for (int u = 0; u < 4; ++u)
        #pragma unroll
        for (int r = 0; r < 8; ++r) s += acc[u][r];
    s += __shfl_xor(s, 16, 32);
    if (lane == 0) wsum[wave] = s;
    __syncthreads();

    if (tid == 0) {
        float rs = 0.0f;
        for (int w = 0; w < NWAVES; ++w) rs += wsum[w];
        partial[b * JSPLIT + js] = rs;
        // every js-block writes the same value for row b (benign identical write)
        npairs[b] = (float)cpos_sh * (float)(N - cpos_sh);
    }
}

__global__ void finalize_kernel(const float* __restrict__ partial,
                                const float* __restrict__ npairs,
                                float* __restrict__ out, int B)
{
    const int lane = threadIdx.x;   // single wave32
    float sum_loss = 0.0f, nvalid = 0.0f;
    for (int b = lane; b < B; b += 32) {
        float rs = 0.0f;
        for (int j = 0; j < JSPLIT; ++j) rs += partial[b * JSPLIT + j];
        float np = npairs[b];
        if (np > 0.0f) { sum_loss += rs / np; nvalid += 1.0f; }
    }
    for (int off = 16; off > 0; off >>= 1) {
        sum_loss += __shfl_down(sum_loss, off, 32);
        nvalid   += __shfl_down(nvalid,   off, 32);
    }
    if (lane == 0) out[0] = (nvalid > 0.0f) ? (sum_loss / nvalid) : 0.0f;
}

extern "C" void kernel_launch(void* const* d_in, const int* in_sizes, int n_in,
                              void* d_out, int out_size, void* d_ws, size_t ws_size,
                              hipStream_t stream)
{
    const float* scores = (const float*)d_in[0];
    const int*   labels = (const int*)d_in[1];
    const int B = in_sizes[0] / N;

    float* partial = (float*)d_ws;            // B * JSPLIT floats
    float* npairs  = partial + B * JSPLIT;    // B floats

    dim3 grid(B, JSPLIT);
    rank_rows_kernel<<<grid, BLK, 0, stream>>>(scores, labels, partial, npairs);
    finalize_kernel<<<1, 32, 0, stream>>>(partial, npairs, (float*)d_out, B);
}